// SDE_35373350650570
// MI455X (gfx1250) — compile-verified
//
#include <hip/hip_runtime.h>
#include <hip/hip_bf16.h>

typedef __attribute__((ext_vector_type(2))) float v2f;
typedef __attribute__((ext_vector_type(8))) float v8f;

#define NPATH   2048
#define DIM     64
#define NSTEPS  500
#define MSTRIDE 66   // padded LDS row stride (floats) -> conflict-free frag loads

// One workgroup = 4 waves = one 16-path tile. Wave w computes output dims
// [16w, 16w+16) of both GEMMs via V_WMMA_F32_16X16X4_F32 (K-loop of 16).
__launch_bounds__(128, 1)
__global__ void sde_euler_wmma(const float* __restrict__ x0,
                               const float* __restrict__ dW,
                               const float* __restrict__ drift,
                               const float* __restrict__ drift_bias,
                               const float* __restrict__ diffusion,
                               const float* __restrict__ diffusion_bias,
                               const float* __restrict__ ts,
                               float* __restrict__ ys) {
    __shared__ float sA[DIM * MSTRIDE];          // drift  (row-major, padded)
    __shared__ float sC[DIM * MSTRIDE];          // diffusion
    __shared__ float sX[2][16 * MSTRIDE];        // x tile, double-buffered
    __shared__ float sTs[NSTEPS + 1];

    const int tid  = threadIdx.x;
    const int lane = tid & 31;
    const int w    = tid >> 5;      // wave id -> n-tile
    const int lr   = lane & 15;
    const int hi   = lane >> 4;     // 0/1 half-wave
    const int p0   = blockIdx.x * 16;

    // ---- one-time init: matrices + x0 tile into LDS, emit ys[:,0,:] ----
    for (int idx = tid; idx < DIM * DIM; idx += 128) {
        int r = idx >> 6, c = idx & 63;
        sA[r * MSTRIDE + c] = drift[idx];
        sC[r * MSTRIDE + c] = diffusion[idx];
    }
    for (int idx = tid; idx < 16 * DIM; idx += 128) {
        int r = idx >> 6, c = idx & 63;
        float v = x0[(size_t)(p0 + r) * DIM + c];
        sX[0][r * MSTRIDE + c] = v;
        ys[((size_t)(p0 + r) * (NSTEPS + 1)) * DIM + c] = v;   // t = 0 snapshot
    }
    for (int i = tid; i <= NSTEPS; i += 128) sTs[i] = ts[i];
    __syncthreads();

    const int   n   = w * 16 + lr;                 // output dim owned by lane
    const float bdb = drift_bias[n];
    const float bsb = diffusion_bias[n];

    // Fragment addressing: A (x-tile): M = lane%16 row, K lo/hi by half-wave.
    // B (= matrixT): kept row-major as drift[n][k], so same pattern with row n.
    const int aoff = lr * MSTRIDE + 2 * hi;        // + 4*kt
    const int boff = n  * MSTRIDE + 2 * hi;        // + 4*kt
    const int uoff = (hi * 8) * MSTRIDE + n;       // + r*MSTRIDE (D-layout path rows)

    for (int t = 0; t < NSTEPS; ++t) {
        const int   cur = t & 1, nxt = cur ^ 1;
        const float dt  = sTs[t + 1] - sTs[t];
        const float sq  = sqrtf(dt);
        const float* xc = sX[cur];

        v8f accD = {};   // drift accumulator  (16x16 D-tile)
        v8f accS = {};   // diffusion accumulator
        #pragma unroll
        for (int kt = 0; kt < 16; ++kt) {
            v2f a  = *(const v2f*)&xc[aoff + kt * 4];
            v2f bD = *(const v2f*)&sA[boff + kt * 4];
            v2f bS = *(const v2f*)&sC[boff + kt * 4];
            accD = __builtin_amdgcn_wmma_f32_16x16x4_f32(
                       false, a, false, bD, (short)0, accD, false, false);
            accS = __builtin_amdgcn_wmma_f32_16x16x4_f32(
                       false, a, false, bS, (short)0, accS, false, false);
        }

        // ---- elementwise Euler update in D layout ----
        #pragma unroll
        for (int r = 0; r < 8; ++r) {
            const int    p   = p0 + r + hi * 8;
            const size_t dwi = ((size_t)p * NSTEPS + t) * DIM + n;
            float dw = dW[dwi];
            float xo = xc[uoff + r * MSTRIDE];
            float xn = xo + dt * (accD[r] + bdb) + (accS[r] + bsb) * (sq * dw);
            sX[nxt][uoff + r * MSTRIDE] = xn;
            ys[((size_t)p * (NSTEPS + 1) + (t + 1)) * DIM + n] = xn;
            if (t + 1 < NSTEPS)
                __builtin_prefetch(&dW[dwi + DIM], 0, 0);   // next step's noise
        }
        __syncthreads();   // publish x_{t+1} to all 4 waves
    }
}

extern "C" void kernel_launch(void* const* d_in, const int* in_sizes, int n_in,
                              void* d_out, int out_size, void* d_ws, size_t ws_size,
                              hipStream_t stream) {
    const float* x0   = (const float*)d_in[0];
    const float* dW   = (const float*)d_in[1];
    const float* drf  = (const float*)d_in[2];
    const float* drfb = (const float*)d_in[3];
    const float* dif  = (const float*)d_in[4];
    const float* difb = (const float*)d_in[5];
    const float* ts   = (const float*)d_in[6];
    float* ys = (float*)d_out;

    dim3 grid(NPATH / 16);   // 128 tiles of 16 paths
    dim3 block(128);         // 4 waves: wave w owns output dims [16w,16w+16)
    sde_euler_wmma<<<grid, block, 0, stream>>>(x0, dW, drf, drfb, dif, difb, ts, ys);
}